// KAN_GNN_88098369175702
// MI455X (gfx1250) — compile-verified
//
#include <hip/hip_runtime.h>

typedef float v2f __attribute__((ext_vector_type(2)));
typedef float v8f __attribute__((ext_vector_type(8)));

#define IN_F  128
#define HID_F 128
#define OUT_F 64
#define KTOT  384   // 3 * IN_F (x, x^2, x^3 stacked)

// ---------------------------------------------------------------- utilities
__global__ void zero_kernel(float* __restrict__ p, long long n) {
  long long i = (long long)blockIdx.x * blockDim.x + threadIdx.x;
  if (i < n) p[i] = 0.0f;
}

// Folded KAN weights, stored K-PAIR-INTERLEAVED so a WMMA B-fragment
// {(kk,o),(kk+1,o)} is one contiguous 8-byte load:
//   wf[((kk>>1)*ncols + o)*2 + (kk&1)] = (p==0 ? w[i,o] : 0) + 0.1*c[i,o,p]
// with kk = p*128 + i.
__global__ void fold_w_kernel(const float* __restrict__ w, const float* __restrict__ c,
                              float* __restrict__ wf, int ncols) {
  int idx = blockIdx.x * blockDim.x + threadIdx.x;
  int total = KTOT * ncols;
  if (idx >= total) return;
  int kk = idx / ncols;
  int o  = idx - kk * ncols;
  int p  = kk >> 7;          // 0,1,2
  int i  = kk & 127;
  float v = 0.1f * c[(i * ncols + o) * 3 + p];
  if (p == 0) v += w[i * ncols + o];
  wf[(size_t)((kk >> 1) * ncols + o) * 2 + (kk & 1)] = v;
}

__global__ void degree_kernel(const int* __restrict__ tgt, float* __restrict__ deg, int nedges) {
  int e = blockIdx.x * blockDim.x + threadIdx.x;
  if (e < nedges) atomicAdd(&deg[tgt[e]], 1.0f);
}

__global__ void invdeg_kernel(float* __restrict__ deg, int n) {
  int i = blockIdx.x * blockDim.x + threadIdx.x;
  if (i < n) deg[i] = 1.0f / fmaxf(deg[i], 1.0f);
}

// ---------------------------------------------------------------- KAN GEMM
// C(nrows x NCT*16) = [A|A^2|A^3](nrows x 384) @ Bf(384 x NCT*16, K-pair packed) + bias
// One wave32 computes a 16-row x (NCT*16)-col strip with v_wmma_f32_16x16x4_f32.
// SCALE_IN folds the mean-aggregation divide into this layer's input load.
template <int NCT, bool RELU, bool SCALE_IN>
__global__ void __launch_bounds__(256)
kan_gemm_kernel(const float* __restrict__ A, const float* __restrict__ scale,
                const float* __restrict__ Bf, const float* __restrict__ bias,
                float* __restrict__ C, int nrows) {
  const int NC   = NCT * 16;
  const int lane = threadIdx.x & 31;
  const int wv   = threadIdx.x >> 5;
  const int rowbase = (blockIdx.x * 8 + wv) * 16;
  if (rowbase >= nrows) return;           // wave-granular exit: EXEC stays all-ones

  const int r  = lane & 15;               // A: row-in-tile ; B/C: col-in-tile
  const int kh = lane >> 4;               // lane-half selects K pairs (ISA 7.12.2)
  const int row = rowbase + r;
  const float sc = SCALE_IN ? scale[row] : 1.0f;
  const v2f* __restrict__ arow = (const v2f*)(A + (size_t)row * IN_F);
  const v2f* __restrict__ Bp   = (const v2f*)Bf;   // (KTOT/2, NC) of K-pairs

  v8f acc[NCT] = {};

  for (int k = 0; k < KTOT; k += 4) {
    // A fragment: vgpr0 -> K = k + 2*kh, vgpr1 -> K = k + 2*kh + 1
    const int kk = k + kh * 2;            // even
    const int p  = kk >> 7;               // basis power segment
    const int i  = kk & 127;              // even => {i, i+1} one contiguous b64
    v2f xv = arow[i >> 1];
    if (SCALE_IN) { xv[0] *= sc; xv[1] *= sc; }
    const float a0 = (p == 0) ? xv[0] : ((p == 1) ? xv[0] * xv[0] : xv[0] * xv[0] * xv[0]);
    const float a1 = (p == 0) ? xv[1] : ((p == 1) ? xv[1] * xv[1] : xv[1] * xv[1] * xv[1]);
    const v2f Af = {a0, a1};

    const v2f* __restrict__ bptr = Bp + (size_t)(kk >> 1) * NC + r;
#pragma unroll
    for (int t = 0; t < NCT; t++) {
      // B fragment {K=kk, K=kk+1} at column t*16+r: single 8-byte load
      const v2f Bfrag = bptr[t * 16];
      acc[t] = __builtin_amdgcn_wmma_f32_16x16x4_f32(
          false, Af, false, Bfrag, (short)0, acc[t], false, false);
    }
  }

  // C layout: vgpr rr, lanes 0-15 -> M=rr, lanes 16-31 -> M=rr+8 ; N = lane&15
#pragma unroll
  for (int t = 0; t < NCT; t++) {
    const int col = t * 16 + r;
    const float bv = bias[col];
#pragma unroll
    for (int rr = 0; rr < 8; rr++) {
      const int orow = rowbase + rr + kh * 8;
      float v = acc[t][rr] + bv;
      if (RELU) v = fmaxf(v, 0.0f);
      C[(size_t)orow * NC + col] = v;
    }
  }
}

// ---------------------------------------------------------------- aggregation
// One lane per (edge, 4-feature group): float4 gather + 4 global f32 atomics.
// h (51/26 MB) is L2-resident (192 MB L2), so gathers and atomics stay on-chip.
template <int F>
__global__ void agg_kernel(const float* __restrict__ h, const int* __restrict__ src,
                           const int* __restrict__ tgt, float* __restrict__ out,
                           int nedges) {
  constexpr int G  = F / 4;
  constexpr int SH = (F == 128) ? 5 : 4;
  long long tid = (long long)blockIdx.x * blockDim.x + threadIdx.x;
  if (tid >= (long long)nedges * G) return;
  const int e = (int)(tid >> SH);
  const int g = (int)(tid & (G - 1));
  const int s = src[e];
  const int t = tgt[e];
  const float4 v = ((const float4*)(h + (size_t)s * F))[g];
  float* op = out + (size_t)t * F + g * 4;
  atomicAdd(op + 0, v.x);
  atomicAdd(op + 1, v.y);
  atomicAdd(op + 2, v.z);
  atomicAdd(op + 3, v.w);
}

// ---------------------------------------------------------------- log-softmax
// One wave32 per node; 64 cols -> 2 per lane; shfl_xor reductions (wave32).
__global__ void logsoftmax_kernel(const float* __restrict__ hin,
                                  const float* __restrict__ invd,
                                  float* __restrict__ out, int nnodes) {
  const int node = (int)(((long long)blockIdx.x * blockDim.x + threadIdx.x) >> 5);
  const int lane = threadIdx.x & 31;
  if (node >= nnodes) return;
  const float sc = invd[node];
  const float* hp = hin + (size_t)node * OUT_F;
  const float y0 = hp[lane] * sc;
  const float y1 = hp[lane + 32] * sc;
  float m = fmaxf(y0, y1);
#pragma unroll
  for (int o = 16; o > 0; o >>= 1) m = fmaxf(m, __shfl_xor(m, o, 32));
  float s = __expf(y0 - m) + __expf(y1 - m);
#pragma unroll
  for (int o = 16; o > 0; o >>= 1) s += __shfl_xor(s, o, 32);
  const float lse = m + __logf(s);
  out[(size_t)node * OUT_F + lane]      = y0 - lse;
  out[(size_t)node * OUT_F + lane + 32] = y1 - lse;
}

// ---------------------------------------------------------------- launcher
extern "C" void kernel_launch(void* const* d_in, const int* in_sizes, int n_in,
                              void* d_out, int out_size, void* d_ws, size_t ws_size,
                              hipStream_t stream) {
  const float* x  = (const float*)d_in[0];
  const int* eidx = (const int*)d_in[1];
  const float* w1 = (const float*)d_in[2];
  const float* b1 = (const float*)d_in[3];
  const float* c1 = (const float*)d_in[4];
  const float* w2 = (const float*)d_in[5];
  const float* b2 = (const float*)d_in[6];
  const float* c2 = (const float*)d_in[7];
  float* out = (float*)d_out;

  const int nnodes = in_sizes[0] / IN_F;
  const int nedges = in_sizes[1] / 2;
  const int* src = eidx;
  const int* tgt = eidx + nedges;

  // workspace layout (floats); bufA reused for H1 then H2, bufB for H1agg then H2agg
  float* ws  = (float*)d_ws;
  float* W1f = ws;                              // 384*128 (K-pair packed)
  float* W2f = W1f + KTOT * HID_F;              // 384*64  (K-pair packed)
  float* invd = W2f + KTOT * OUT_F;             // nnodes (deg -> inv_deg in place)
  size_t bufoff = (size_t)(KTOT * HID_F + KTOT * OUT_F) + (size_t)nnodes;
  bufoff = (bufoff + 255) & ~(size_t)255;
  float* bufA = ws + bufoff;                    // nnodes * 128
  float* bufB = bufA + (size_t)nnodes * HID_F;  // nnodes * 128

  // 1) fold KAN weights into single 384-K GEMM operands (packed for b64 fragments)
  fold_w_kernel<<<(KTOT * HID_F + 255) / 256, 256, 0, stream>>>(w1, c1, W1f, HID_F);
  fold_w_kernel<<<(KTOT * OUT_F + 255) / 256, 256, 0, stream>>>(w2, c2, W2f, OUT_F);

  // 2) degrees -> inverse degrees (clamped at 1)
  zero_kernel<<<(nnodes + 255) / 256, 256, 0, stream>>>(invd, nnodes);
  degree_kernel<<<(nedges + 255) / 256, 256, 0, stream>>>(tgt, invd, nedges);
  invdeg_kernel<<<(nnodes + 255) / 256, 256, 0, stream>>>(invd, nnodes);

  // 3) layer 1: H1 = relu(KAN1(x)) -> bufA
  const int nwaves  = (nnodes + 15) / 16;
  const int gblocks = (nwaves + 7) / 8;
  kan_gemm_kernel<8, true, false><<<gblocks, 256, 0, stream>>>(x, nullptr, W1f, b1, bufA, nnodes);

  // 4) aggregate H1 -> bufB (sum; divide folded into layer-2 load)
  long long n1 = (long long)nnodes * HID_F;
  zero_kernel<<<(int)((n1 + 255) / 256), 256, 0, stream>>>(bufB, n1);
  long long t1 = (long long)nedges * (HID_F / 4);
  agg_kernel<HID_F><<<(int)((t1 + 255) / 256), 256, 0, stream>>>(bufA, src, tgt, bufB, nedges);

  // 5) layer 2: H2 = KAN2(bufB * inv_deg) -> bufA (64 cols)
  kan_gemm_kernel<4, false, true><<<gblocks, 256, 0, stream>>>(bufB, invd, W2f, b2, bufA, nnodes);

  // 6) aggregate H2 -> bufB
  long long n2 = (long long)nnodes * OUT_F;
  zero_kernel<<<(int)((n2 + 255) / 256), 256, 0, stream>>>(bufB, n2);
  long long t2 = (long long)nedges * (OUT_F / 4);
  agg_kernel<OUT_F><<<(int)((t2 + 255) / 256), 256, 0, stream>>>(bufA, src, tgt, bufB, nedges);

  // 7) log-softmax(bufB * inv_deg) -> out
  logsoftmax_kernel<<<(nnodes + 7) / 8, 256, 0, stream>>>(bufB, invd, out, nnodes);
}